// Model_57415122812994
// MI455X (gfx1250) — compile-verified
//
#include <hip/hip_runtime.h>

// Problem constants (match reference)
#define B_ROWS 16384
#define IN_F   2048
#define OUT_F  8192

#define NCHUNK 64                       // row-chunks for W column-sum
#define ROWS_PER_CHUNK (OUT_F / NCHUNK) // 128
#define KSPLIT 4                        // K-split for the WMMA GEMV
#define KCHUNK (IN_F / KSPLIT)          // 512

typedef __attribute__((ext_vector_type(2))) float v2f;
typedef __attribute__((ext_vector_type(8))) float v8f;

// Workspace layout (float offsets):
//  [0, NCHUNK*IN_F)        : W column-sum partials (64 x 2048)
//  [WSUM_OFF, +IN_F)       : wsum (2048)
//  [BSUM_OFF]              : bsum (padded)
//  [DP_OFF, +B_ROWS*KSPLIT): GEMV dot partials (16384 x 4)
#define WSUM_OFF (NCHUNK * IN_F)
#define BSUM_OFF (WSUM_OFF + IN_F)
#define DP_OFF   (BSUM_OFF + 64)

// --- Kernel 1: partial column sums of W (deterministic, no atomics) -------
__global__ void k_colsum_partial(const float* __restrict__ W, float* __restrict__ ws) {
    int col   = blockIdx.x * blockDim.x + threadIdx.x;   // 0..2047
    int chunk = blockIdx.y;                              // 0..63
    const float* p = W + (size_t)chunk * ROWS_PER_CHUNK * IN_F + col;
    float s = 0.0f;
#pragma unroll 8
    for (int r = 0; r < ROWS_PER_CHUNK; ++r)
        s += p[(size_t)r * IN_F];
    ws[chunk * IN_F + col] = s;
}

// --- Kernel 2: fold partials -> wsum --------------------------------------
__global__ void k_colsum_final(float* __restrict__ ws) {
    int col = blockIdx.x * blockDim.x + threadIdx.x;     // 0..2047
    float s = 0.0f;
#pragma unroll
    for (int c = 0; c < NCHUNK; ++c)
        s += ws[c * IN_F + col];
    ws[WSUM_OFF + col] = s;
}

// --- Kernel 3: bsum = sum(b) ----------------------------------------------
__global__ void k_bias_sum(const float* __restrict__ b, float* __restrict__ ws) {
    __shared__ float red[256];
    float s = 0.0f;
    for (int i = threadIdx.x; i < OUT_F; i += 256)
        s += b[i];
    red[threadIdx.x] = s;
    __syncthreads();
    for (int off = 128; off > 0; off >>= 1) {
        if ((int)threadIdx.x < off) red[threadIdx.x] += red[threadIdx.x + off];
        __syncthreads();
    }
    if (threadIdx.x == 0) ws[BSUM_OFF] = red[0];
}

// --- Kernel 4: GEMV via V_WMMA_F32_16X16X4_F32 ----------------------------
// One wave = one (16-row tile, K-chunk) unit. A = x tile (16x4), B = wsum
// broadcast to all 16 columns, C accumulates in f32. EXEC is all-1s (full
// blocks, no divergence before the WMMA).
__global__ void k_gemv_wmma(const float* __restrict__ x,
                            const float* __restrict__ ws,
                            float* __restrict__ dp) {
    const float* wsum = ws + WSUM_OFF;
    int lane    = threadIdx.x & 31;
    int wid     = (blockIdx.x * blockDim.x + threadIdx.x) >> 5;
    int tile    = wid >> 2;            // / KSPLIT
    int ks      = wid & (KSPLIT - 1);
    int halfsel = lane >> 4;           // 0: K={0,1}, 1: K={2,3}
    int row     = tile * 16 + (lane & 15);
    const float* xrow = x + (size_t)row * IN_F;

    int kbeg = ks * KCHUNK;
    int kend = kbeg + KCHUNK;
    v8f c = {0.f, 0.f, 0.f, 0.f, 0.f, 0.f, 0.f, 0.f};

#pragma unroll 4
    for (int k = kbeg; k < kend; k += 4) {
        int off = k + halfsel * 2;
        v2f a = *(const v2f*)(xrow + off);   // A: row M=lane%16, K pair
        v2f b = *(const v2f*)(wsum + off);   // B: same value in all 16 cols
        c = __builtin_amdgcn_wmma_f32_16x16x4_f32(
                /*neg_a=*/false, a, /*neg_b=*/false, b,
                /*c_mod=*/(short)0, c, /*reuse_a=*/false, /*reuse_b=*/false);
    }

    // C/D layout: VGPR r -> (M=r, N=lane) for lanes 0-15, (M=8+r) for 16-31.
    // All columns identical, so lanes 0 and 16 each hold 8 row results.
    if ((lane & 15) == 0) {
        int base = tile * 16 + halfsel * 8;
        dp[(size_t)(base + 0) * KSPLIT + ks] = c[0];
        dp[(size_t)(base + 1) * KSPLIT + ks] = c[1];
        dp[(size_t)(base + 2) * KSPLIT + ks] = c[2];
        dp[(size_t)(base + 3) * KSPLIT + ks] = c[3];
        dp[(size_t)(base + 4) * KSPLIT + ks] = c[4];
        dp[(size_t)(base + 5) * KSPLIT + ks] = c[5];
        dp[(size_t)(base + 6) * KSPLIT + ks] = c[6];
        dp[(size_t)(base + 7) * KSPLIT + ks] = c[7];
    }
}

// --- Kernel 5: fold K-split partials + bsum -> out ------------------------
__global__ void k_finalize(const float* __restrict__ ws, float* __restrict__ out) {
    int i = blockIdx.x * blockDim.x + threadIdx.x;       // 0..16383
    const float* dp = ws + DP_OFF;
    float s = dp[(size_t)i * KSPLIT + 0] + dp[(size_t)i * KSPLIT + 1]
            + dp[(size_t)i * KSPLIT + 2] + dp[(size_t)i * KSPLIT + 3];
    out[i] = s + ws[BSUM_OFF];
}

extern "C" void kernel_launch(void* const* d_in, const int* in_sizes, int n_in,
                              void* d_out, int out_size, void* d_ws, size_t ws_size,
                              hipStream_t stream) {
    const float* x = (const float*)d_in[0];   // (16384, 2048)
    const float* W = (const float*)d_in[1];   // (8192, 2048)
    const float* b = (const float*)d_in[2];   // (8192,)
    float* out = (float*)d_out;               // (16384, 1)
    float* ws  = (float*)d_ws;

    // 1) W column-sum partials: grid (2048/256, 64)
    dim3 g1(IN_F / 256, NCHUNK);
    k_colsum_partial<<<g1, 256, 0, stream>>>(W, ws);

    // 2) fold -> wsum
    k_colsum_final<<<IN_F / 256, 256, 0, stream>>>(ws);

    // 3) bsum
    k_bias_sum<<<1, 256, 0, stream>>>(b, ws);

    // 4) WMMA GEMV: (16384/16) tiles * KSPLIT waves = 4096 waves, 8 waves/block
    int total_waves  = (B_ROWS / 16) * KSPLIT;
    int blocks_gemv  = total_waves / 8;       // 256 threads/block
    k_gemv_wmma<<<blocks_gemv, 256, 0, stream>>>(x, ws, ws + DP_OFF);

    // 5) finalize
    k_finalize<<<B_ROWS / 256, 256, 0, stream>>>(ws, out);
}